// CubicShapeFunction_85280870629635
// MI455X (gfx1250) — compile-verified
//
#include <hip/hip_runtime.h>
#include <stdint.h>

#define TPB 256
#define PPB 16   // particles per block: 16 threads/particle, 4 stencil points/thread

typedef float        v4f      __attribute__((ext_vector_type(4)));
typedef unsigned int tdm_g0_t __attribute__((ext_vector_type(4)));
typedef int          tdm_g1_t __attribute__((ext_vector_type(8)));
typedef int          tdm_v4i  __attribute__((ext_vector_type(4)));
typedef int          tdm_v8i  __attribute__((ext_vector_type(8)));

__global__ __launch_bounds__(TPB) void CubicShapeFunction_kernel(
    const float* __restrict__ pos,   // [n,3]
    float* __restrict__ outS,        // [n,64]
    float* __restrict__ outG,        // [n,64,3]
    int n)
{
    __shared__ __align__(16) float sPos[PPB * 3];      // staged positions (48 floats)
    __shared__ __align__(16) float sB[PPB][3][4];      // basis
    __shared__ __align__(16) float sD[PPB][3][4];      // d(basis) * h

    const int t     = (int)threadIdx.x;
    const int pbase = (int)blockIdx.x * PPB;
    const int prem  = n - pbase;                        // particles left for this block

    // ---- TDM: async DMA of this block's positions (<=192B) into LDS ----
    // 1D tensor descriptor: data_size=4B, tensor_dim0 clamped to remaining
    // elements (OOB reads return zero per ISA), tile 48x1, stride 48.
    if ((t >> 5) == 0) {  // wave 0 only
        const uint64_t ga    = (uint64_t)(uintptr_t)pos + (uint64_t)pbase * 12u;
        const uint32_t lds   = (uint32_t)(uintptr_t)(&sPos[0]);
        const uint32_t ndim0 = (uint32_t)((prem < PPB ? prem : PPB) * 3);

        tdm_g0_t g0;
        g0[0] = 1u;                                                 // count=1, user desc
        g0[1] = lds;                                                // lds_addr (bytes)
        g0[2] = (uint32_t)ga;                                       // global_addr[31:0]
        g0[3] = (uint32_t)((ga >> 32) & 0x01FFFFFFu) | (2u << 30);  // ga[56:32], type=2

        tdm_g1_t g1;
        g1[0] = (int)(2u << 16);         // workgroup_mask=0, data_size=2 -> 4 bytes
        g1[1] = (int)(ndim0 << 16);      // tensor_dim0[15:0] @ bits[63:48]
        g1[2] = (int)(1u << 16);         // tensor_dim0 hi=0, tensor_dim1=1
        g1[3] = (int)(48u << 16);        // tensor_dim1 hi=0, tile_dim0=48
        g1[4] = 1;                       // tile_dim1=1, tile_dim2=0
        g1[5] = 48;                      // tensor_dim0_stride lo
        g1[6] = 0;                       // stride hi / tensor_dim1_stride lo
        g1[7] = 0;

        tdm_v4i gz4 = {0, 0, 0, 0};
        tdm_v8i gz8 = {0, 0, 0, 0, 0, 0, 0, 0};
        __builtin_amdgcn_tensor_load_to_lds(g0, g1, gz4, gz4, gz8, 0);
        __builtin_amdgcn_s_wait_tensorcnt(0);
    }
    __syncthreads();

    // ---- Phase 1: 12 threads/particle evaluate the 1D splines -> LDS ----
    // dist = frac + 1 - o with frac in [0,1): branch is fixed per offset o.
    if (t < PPB * 12) {
        const int p = t / 12;
        const int r = t - p * 12;
        const int d = r >> 2;            // dim 0..2
        const int o = r & 3;             // stencil offset 0..3
        if (p < prem) {
            const float rel = sPos[p * 3 + d] * 20.0f;       // INV_CELL
            const float x   = (rel - floorf(rel)) + 1.0f - (float)o;
            float b, db;
            if (o == 0) {                 // x in [1,2)
                b  = ((-(1.0f/6.0f) * x + 1.0f) * x - 2.0f) * x + (4.0f/3.0f);
                db = 20.0f * ((-0.5f * x + 2.0f) * x - 2.0f);
            } else if (o == 1) {          // x in [0,1)
                b  = (0.5f * x - 1.0f) * x * x + (2.0f/3.0f);
                db = 20.0f * ((1.5f * x - 2.0f) * x);
            } else if (o == 2) {          // x in [-1,0)
                b  = (-0.5f * x - 1.0f) * x * x + (2.0f/3.0f);
                db = 20.0f * ((-1.5f * x - 2.0f) * x);
            } else {                      // x in [-2,-1)
                b  = (((1.0f/6.0f) * x + 1.0f) * x + 2.0f) * x + (4.0f/3.0f);
                db = 20.0f * ((0.5f * x + 2.0f) * x + 2.0f);
            }
            sB[p][d][o] = b;
            sD[p][d][o] = db;
        }
    }
    __syncthreads();

    // ---- Phase 2: combine + perfectly coalesced b128 NT stores ----
    const int p  = t >> 4;               // local particle
    const int q  = t & 15;               // stencil group: j = q*4 + iz
    const int ix = q >> 2;
    const int iy = q & 3;

    const float bx  = sB[p][0][ix];
    const float dbx = sD[p][0][ix];
    const float by  = sB[p][1][iy];
    const float dby = sD[p][1][iy];
    const v4f   bz  = *(const v4f*)(&sB[p][2][0]);   // ds_load_b128
    const v4f   dbz = *(const v4f*)(&sD[p][2][0]);

    const float bxy = bx * by;
    const float gx  = dbx * by;          // -> grad x coefficient
    const float gy  = bx * dby;          // -> grad y coefficient

    v4f s;
    s.x = bxy * bz.x; s.y = bxy * bz.y; s.z = bxy * bz.z; s.w = bxy * bz.w;

    // 12 grad floats for the 4 stencil points, packed as 3x v4f
    v4f ga, gb, gc;
    ga.x = gx  * bz.x;  ga.y = gy  * bz.x;  ga.z = bxy * dbz.x;  ga.w = gx * bz.y;
    gb.x = gy  * bz.y;  gb.y = bxy * dbz.y; gb.z = gx  * bz.z;   gb.w = gy * bz.z;
    gc.x = bxy * dbz.z; gc.y = gx  * bz.w;  gc.z = gy  * bz.w;   gc.w = bxy * dbz.w;

    if (p < prem) {
        // shapef index: (pbase+p)*64 + q*4 == pbase*64 + 4*t  -> lane-contiguous b128
        v4f* ps = (v4f*)(outS + (size_t)pbase * 64 + (size_t)t * 4);
        __builtin_nontemporal_store(s, ps);
        // grad index: ((pbase+p)*64 + q*4)*3 == pbase*192 + 12*t -> 3 contiguous b128
        v4f* pg = (v4f*)(outG + (size_t)pbase * 192 + (size_t)t * 12);
        __builtin_nontemporal_store(ga, pg);
        __builtin_nontemporal_store(gb, pg + 1);
        __builtin_nontemporal_store(gc, pg + 2);
    }
}

extern "C" void kernel_launch(void* const* d_in, const int* in_sizes, int n_in,
                              void* d_out, int out_size, void* d_ws, size_t ws_size,
                              hipStream_t stream) {
    const float* pos = (const float*)d_in[0];
    const int n = in_sizes[0] / 3;                 // 500,000
    float* outS = (float*)d_out;                   // [n,64]
    float* outG = outS + (size_t)n * 64;           // [n,64,3]
    if (n <= 0) return;
    const int blocks = (n + PPB - 1) / PPB;
    CubicShapeFunction_kernel<<<dim3(blocks), dim3(TPB), 0, stream>>>(pos, outS, outG, n);
}